// ScaleSAGE_85023172592273
// MI455X (gfx1250) — compile-verified
//
#include <hip/hip_runtime.h>
#include <hip/hip_bf16.h>
#include <math.h>

typedef __attribute__((ext_vector_type(2))) float v2f;
typedef __attribute__((ext_vector_type(8))) float v8f;

#define N_TOTAL 500000
#define N0      500000

// ---------------------------------------------------------------------------
// Edge scatter: s[dst] += x[src] * keep ; cnt[dst] += keep
// keep = !cached[n_id[dst]] (or 1 when cached == nullptr).
// block = (C, 256/C): each threadIdx.y handles one edge, lanes cover channels.
// Destination rows live in L2 (<=51MB), so f32 atomics resolve on-chip.
// ---------------------------------------------------------------------------
__global__ void sage_scatter(const float* __restrict__ x,
                             const int* __restrict__ src,
                             const int* __restrict__ dst,
                             const int* __restrict__ n_id,
                             const unsigned char* __restrict__ cached,
                             float* __restrict__ s,
                             float* __restrict__ cnt,
                             int E, int C) {
    int e = blockIdx.x * blockDim.y + threadIdx.y;
    if (e >= E) return;
    int d = dst[e];
    if (cached != nullptr) {
        if (cached[n_id[d]] != 0) return;   // keep == 0: contributes nothing
    }
    int sr = src[e];
    int c = threadIdx.x;
    atomicAdd(&s[(size_t)d * C + c], x[(size_t)sr * C + c]);
    if (c == 0) atomicAdd(&cnt[d], 1.0f);
}

// ---------------------------------------------------------------------------
// Fused SAGE GEMM: out = relu?( (s/max(cnt,1)) @ Wl + xin @ Wr + bias ) with
// optional cached-embedding substitution.  One wave32 per 16x16 output tile,
// V_WMMA_F32_16X16X4_F32, K-loop in steps of 4.  All M,N multiples of 16.
//
// A (16x4 f32) per-lane: m = lane&15, kb = 2*(lane>>4), elems = A[m][kb+i]
// B (4x16 f32) per-lane: n = lane&15, kb = 2*(lane>>4), elems = B[kb+i][n]
// C/D (16x16 f32): vgpr r -> row = r + 8*(lane>>4), col = lane&15
// ---------------------------------------------------------------------------
__global__ void sage_gemm_wmma(const float* __restrict__ s,
                               const float* __restrict__ cnt,
                               const float* __restrict__ xin,
                               const float* __restrict__ Wl,
                               const float* __restrict__ Wr,
                               const float* __restrict__ bias,
                               float* __restrict__ out,
                               const int* __restrict__ n_id,
                               const unsigned char* __restrict__ cached,
                               const float* __restrict__ emb,
                               int K, int N, int relu_emb) {
    const int lane = threadIdx.x;        // 0..31
    const int half = lane >> 4;          // 0 or 1
    const int lm   = lane & 15;

    const int mrow = blockIdx.x * 16 + lm;   // A-row this lane feeds
    const int ncol = blockIdx.y * 16 + lm;   // B-col this lane feeds
    const int kb   = half * 2;

    const float inv = 1.0f / fmaxf(cnt[mrow], 1.0f);
    const float* __restrict__ srow = s   + (size_t)mrow * K;
    const float* __restrict__ xrow = xin + (size_t)mrow * K;

    v8f acc = {};
    for (int k0 = 0; k0 < K; k0 += 4) {
        const int k = k0 + kb;
        v2f a1, a2, b1, b2;
        a1[0] = srow[k]     * inv;
        a1[1] = srow[k + 1] * inv;
        a2[0] = xrow[k];
        a2[1] = xrow[k + 1];
        b1[0] = Wl[(size_t)k * N + ncol];
        b1[1] = Wl[(size_t)(k + 1) * N + ncol];
        b2[0] = Wr[(size_t)k * N + ncol];
        b2[1] = Wr[(size_t)(k + 1) * N + ncol];
        acc = __builtin_amdgcn_wmma_f32_16x16x4_f32(
            false, a1, false, b1, (short)0, acc, false, false);
        acc = __builtin_amdgcn_wmma_f32_16x16x4_f32(
            false, a2, false, b2, (short)0, acc, false, false);
    }

    const float bv = bias[ncol];
#pragma unroll
    for (int r = 0; r < 8; ++r) {
        const int orow = blockIdx.x * 16 + r + 8 * half;
        float v = acc[r] + bv;
        if (relu_emb) {
            v = fmaxf(v, 0.0f);
            const int nid = n_id[orow];
            if (cached[nid]) v = emb[(size_t)nid * N + ncol];  // N == 256 here
        }
        out[(size_t)orow * N + ncol] = v;
    }
}

// ---------------------------------------------------------------------------
// In-place row-wise log_softmax, C = 64: one wave per row, 2 values per lane,
// wave32 shuffle reductions.
// ---------------------------------------------------------------------------
__global__ void log_softmax64(float* __restrict__ out, int M) {
    const int lane = threadIdx.x & 31;
    const int wid  = threadIdx.x >> 5;
    const int row  = blockIdx.x * (blockDim.x >> 5) + wid;
    if (row >= M) return;
    float* p = out + (size_t)row * 64;
    float v0 = p[lane];
    float v1 = p[lane + 32];
    float m = fmaxf(v0, v1);
#pragma unroll
    for (int o = 16; o > 0; o >>= 1) m = fmaxf(m, __shfl_xor(m, o, 32));
    float sum = __expf(v0 - m) + __expf(v1 - m);
#pragma unroll
    for (int o = 16; o > 0; o >>= 1) sum += __shfl_xor(sum, o, 32);
    const float lse = m + __logf(sum);
    p[lane]      = v0 - lse;
    p[lane + 32] = v1 - lse;
}

extern "C" void kernel_launch(void* const* d_in, const int* in_sizes, int n_in,
                              void* d_out, int out_size, void* d_ws, size_t ws_size,
                              hipStream_t stream) {
    const float* x    = (const float*)d_in[0];
    const float* Wl0  = (const float*)d_in[1];
    const float* Wr0  = (const float*)d_in[2];
    const float* b0   = (const float*)d_in[3];
    const float* Wl1  = (const float*)d_in[4];
    const float* Wr1  = (const float*)d_in[5];
    const float* b1   = (const float*)d_in[6];
    const float* Wl2  = (const float*)d_in[7];
    const float* Wr2  = (const float*)d_in[8];
    const float* b2   = (const float*)d_in[9];
    const float* emb0 = (const float*)d_in[10];
    const float* emb1 = (const float*)d_in[11];
    const int*   n_id = (const int*)d_in[12];
    const int*   src0 = (const int*)d_in[13];
    const int*   dst0 = (const int*)d_in[14];
    const int*   src1 = (const int*)d_in[15];
    const int*   dst1 = (const int*)d_in[16];
    const int*   src2 = (const int*)d_in[17];
    const int*   dst2 = (const int*)d_in[18];
    const unsigned char* cached0 = (const unsigned char*)d_in[19];
    const unsigned char* cached1 = (const unsigned char*)d_in[20];
    float* outp = (float*)d_out;

    // Workspace layout (floats). S is reused for all three layers' scatter.
    float* S   = (float*)d_ws;            // 100000*128 = 12,800,000
    float* CNT = S   + 12800000;          // 100,000
    float* H1  = CNT + 100000;            // 100000*256 = 25,600,000
    float* H2  = H1  + 25600000;          // 20000*256  =  5,120,000
    (void)ws_size; (void)n_in; (void)in_sizes; (void)out_size;

    // ---- Layer 0: (500000,128) -> (100000,256) -------------------------
    hipMemsetAsync(S,   0, (size_t)100000 * 128 * sizeof(float), stream);
    hipMemsetAsync(CNT, 0, (size_t)100000 * sizeof(float), stream);
    {
        dim3 blk(128, 2);                              // 2 edges per block
        int grid = (1000000 + 1) / 2;
        sage_scatter<<<grid, blk, 0, stream>>>(x, src0, dst0, n_id, cached0,
                                               S, CNT, 1000000, 128);
        dim3 g(100000 / 16, 256 / 16);
        sage_gemm_wmma<<<g, 32, 0, stream>>>(S, CNT, x, Wl0, Wr0, b0, H1,
                                             n_id, cached0, emb0, 128, 256, 1);
    }

    // ---- Layer 1: (100000,256) -> (20000,256) --------------------------
    hipMemsetAsync(S,   0, (size_t)20000 * 256 * sizeof(float), stream);
    hipMemsetAsync(CNT, 0, (size_t)20000 * sizeof(float), stream);
    {
        dim3 blk(256, 1);
        sage_scatter<<<200000, blk, 0, stream>>>(H1, src1, dst1, n_id, cached1,
                                                 S, CNT, 200000, 256);
        dim3 g(20000 / 16, 256 / 16);
        sage_gemm_wmma<<<g, 32, 0, stream>>>(S, CNT, H1, Wl1, Wr1, b1, H2,
                                             n_id, cached1, emb1, 256, 256, 1);
    }

    // ---- Layer 2: (20000,256) -> (4096,64), straight into d_out --------
    hipMemsetAsync(S,   0, (size_t)4096 * 256 * sizeof(float), stream);
    hipMemsetAsync(CNT, 0, (size_t)4096 * sizeof(float), stream);
    {
        dim3 blk(256, 1);
        sage_scatter<<<40960, blk, 0, stream>>>(H2, src2, dst2, n_id, nullptr,
                                                S, CNT, 40960, 256);
        dim3 g(4096 / 16, 64 / 16);
        sage_gemm_wmma<<<g, 32, 0, stream>>>(S, CNT, H2, Wl2, Wr2, b2, outp,
                                             n_id, nullptr, nullptr, 256, 64, 0);
    }

    // ---- log_softmax over 64 classes, in place -------------------------
    log_softmax64<<<4096 / 8, 256, 0, stream>>>(outp, 4096);
}